// OptimalTexture_66821101191724
// MI455X (gfx1250) — compile-verified
//
#include <hip/hip_runtime.h>
#include <hip/hip_bf16.h>
#include <stdint.h>

#define CCH 256               // channel count C
#define EPSR 1e-5f            // covariance regularizer

typedef __attribute__((ext_vector_type(2))) float v2f;
typedef __attribute__((ext_vector_type(8))) float v8f;
typedef __attribute__((ext_vector_type(4))) int   i32x4;

// pointer flavors for the async load-to-LDS builtin (param 1 is
// 'int __vector(4) __device__ *', i.e. addrspace(1) int4*)
typedef __attribute__((address_space(1))) i32x4 gbl_i32x4;
typedef __attribute__((address_space(3))) i32x4 lds_i32x4;

// ---- CDNA5 async load-to-LDS availability probes -------------------------
#if defined(__has_builtin)
#if __has_builtin(__builtin_amdgcn_global_load_async_to_lds_b128)
#define HAVE_ASYNC_LDS 1
#endif
#if __has_builtin(__builtin_amdgcn_s_wait_asynccnt)
#define HAVE_WAIT_ASYNC 1
#endif
#endif
#ifndef HAVE_ASYNC_LDS
#define HAVE_ASYNC_LDS 0
#endif
#ifndef HAVE_WAIT_ASYNC
#define HAVE_WAIT_ASYNC 0
#endif

// ---------------------------------------------------------------------------
// CDNA5 fp32 WMMA: D(16x16 f32) = A(16x4 f32) * B(4x16 f32) + C
// ---------------------------------------------------------------------------
static __device__ __forceinline__ v8f wmma4(v2f a, v2f b, v8f c) {
  return __builtin_amdgcn_wmma_f32_16x16x4_f32(false, a, false, b,
                                               (short)0, c, false, false);
}

// ---------------------------------------------------------------------------
// trivial utility kernels
// ---------------------------------------------------------------------------
__global__ void k_copy(float* __restrict__ dst, const float* __restrict__ src, int n) {
  int i = blockIdx.x * blockDim.x + threadIdx.x;
  if (i < n) dst[i] = src[i];
}

__global__ void k_zero(float* __restrict__ p, int n) {
  int i = blockIdx.x * blockDim.x + threadIdx.x;
  if (i < n) p[i] = 0.0f;
}

__global__ void k_scale(float* __restrict__ p, float s, int n) {
  int i = blockIdx.x * blockDim.x + threadIdx.x;
  if (i < n) p[i] *= s;
}

// column sums of an [N x 256] matrix; block reduces 256 rows, thread's channel
// fixed (coalesced), one f32 atomic per thread at the end.
__global__ void k_colsum(const float* __restrict__ src, float* __restrict__ mu) {
  int c = threadIdx.x;  // 256 threads
  const float* p = src + (size_t)blockIdx.x * 256u * CCH + c;
  float s = 0.0f;
#pragma unroll 8
  for (int r = 0; r < 256; ++r) s += p[(size_t)r * CCH];
  atomicAdd(&mu[c], s);
}

// ---------------------------------------------------------------------------
// Gaussian matrix generation (hash + Box-Muller), deterministic in (seed, idx)
// ---------------------------------------------------------------------------
static __device__ __forceinline__ uint32_t hash_u32(uint32_t x) {
  x ^= x >> 16; x *= 0x7feb352du;
  x ^= x >> 15; x *= 0x846ca68bu;
  x ^= x >> 16;
  return x;
}

__global__ void k_gauss(float* __restrict__ out, unsigned seed) {
  unsigned idx = blockIdx.x * 256u + threadIdx.x;   // 65536 elements
  uint32_t h1 = hash_u32(idx * 2u + seed);
  uint32_t h2 = hash_u32(idx * 2u + 1u + seed * 0x85ebca6bu);
  float u1 = ((float)h1 + 1.0f) * (1.0f / 4294967296.0f);
  float u2 = (float)h2 * (1.0f / 4294967296.0f);
  float r = sqrtf(-2.0f * logf(u1));
  out[idx] = r * cosf(6.28318530718f * u2);
}

// ---------------------------------------------------------------------------
// Modified Gram-Schmidt QR of a 256x256 matrix in LDS; writes Q (Haar rotation)
// ---------------------------------------------------------------------------
#define QS 257
__global__ void k_qr(const float* __restrict__ A, float* __restrict__ Q) {
  extern __shared__ float lds[];
  float* v = lds;                // 256 columns x QS
  float* red = lds + 256 * QS;   // 256-float reduction scratch
  int t = threadIdx.x;
  for (int i = 0; i < 256; ++i) v[t * QS + i] = A[(size_t)i * CCH + t];
  __syncthreads();
  for (int k = 0; k < 256; ++k) {
    float x = v[k * QS + t];
    red[t] = x * x;
    __syncthreads();
    for (int off = 128; off > 0; off >>= 1) {
      if (t < off) red[t] += red[t + off];
      __syncthreads();
    }
    float inv = rsqrtf(red[0] + 1e-30f);
    __syncthreads();
    v[k * QS + t] *= inv;
    __syncthreads();
    if (t > k) {
      float r = 0.0f;
      for (int i = 0; i < 256; ++i) r += v[k * QS + i] * v[t * QS + i];
      for (int i = 0; i < 256; ++i) v[t * QS + i] -= r * v[k * QS + i];
    }
    __syncthreads();
  }
  for (int i = 0; i < 256; ++i) Q[(size_t)i * CCH + t] = v[t * QS + i];
}

// ---------------------------------------------------------------------------
// in-place Cholesky of a 256x256 SPD matrix (lower L, upper zeroed)
// ---------------------------------------------------------------------------
__global__ void k_chol(float* __restrict__ M) {
  extern __shared__ float a[];
  int t = threadIdx.x;
  for (int i = t; i < 65536; i += 256) a[(i >> 8) * 257 + (i & 255)] = M[i];
  __syncthreads();
  for (int k = 0; k < 256; ++k) {
    if (t == 0) a[k * 257 + k] = sqrtf(fmaxf(a[k * 257 + k], 1e-30f));
    __syncthreads();
    float dk = a[k * 257 + k];
    if (t > k) a[t * 257 + k] /= dk;
    __syncthreads();
    if (t > k) {
      float lik = a[t * 257 + k];
      for (int j = k + 1; j <= t; ++j) a[t * 257 + j] -= lik * a[j * 257 + k];
    }
    __syncthreads();
  }
  for (int i = t; i < 65536; i += 256) {
    int r = i >> 8, c = i & 255;
    M[i] = (c <= r) ? a[r * 257 + c] : 0.0f;
  }
}

// ---------------------------------------------------------------------------
// X = Ls * inv(Lt)   (solve X * Lt = Ls, Lt lower-triangular)
// ---------------------------------------------------------------------------
__global__ void k_trisolve(const float* __restrict__ Lt,
                           const float* __restrict__ Ls,
                           float* __restrict__ X) {
  extern __shared__ float x[];
  int t = threadIdx.x;
  for (int i = t; i < 65536; i += 256) x[(i >> 8) * 257 + (i & 255)] = Ls[i];
  __syncthreads();
  for (int j = 255; j >= 0; --j) {
    float s = x[t * 257 + j];
    for (int k = j + 1; k < 256; ++k) s -= x[t * 257 + k] * Lt[(size_t)k * CCH + j];
    x[t * 257 + j] = s / Lt[(size_t)j * CCH + j];
  }
  __syncthreads();
  for (int i = t; i < 65536; i += 256) X[i] = x[(i >> 8) * 257 + (i & 255)];
}

// ---------------------------------------------------------------------------
// generic small 256x256x256 matmul with transpose flags + optional eps*I
// ---------------------------------------------------------------------------
__global__ void k_mm256(const float* __restrict__ A, const float* __restrict__ B,
                        float* __restrict__ O, int ta, int tb, float epsd) {
  int i = blockIdx.x, j = threadIdx.x;
  float s = 0.0f;
  for (int k = 0; k < 256; ++k) {
    float av = ta ? A[k * CCH + i] : A[i * CCH + k];
    float bv = tb ? B[j * CCH + k] : B[k * CCH + j];
    s = fmaf(av, bv, s);
  }
  if (i == j) s += epsd;
  O[i * CCH + j] = s;
}

// ---------------------------------------------------------------------------
// Cov += ct^T @ ct with on-the-fly centering.  One wave per (16x16 tile,
// K-slice); WMMA f32 16x16x4 inner loop; split-K combined via f32 atomics.
// ---------------------------------------------------------------------------
__global__ void k_cov(const float* __restrict__ src, const float* __restrict__ mu,
                      float* __restrict__ out, int chunk) {
  int l = threadIdx.x;
  int m = l & 15, h = l >> 4;
  int ci = blockIdx.x * 16, cj = blockIdx.y * 16;
  size_t k0 = (size_t)blockIdx.z * (size_t)chunk;
  float mua = mu[ci + m];
  float mub = mu[cj + m];
  const float* pa = src + k0 * CCH + ci + m;
  const float* pb = src + k0 * CCH + cj + m;
  v8f acc = {};
  for (int k = 0; k < chunk; k += 4) {
    size_t off = (size_t)(k + 2 * h) * CCH;
    v2f a, b;
    a.x = pa[off] - mua;       // A^T[m][k+2h]   = ct[k+2h][ci+m]
    a.y = pa[off + CCH] - mua;
    b.x = pb[off] - mub;       // B[k+2h][n=m]
    b.y = pb[off + CCH] - mub;
    acc = wmma4(a, b, acc);
  }
#pragma unroll
  for (int v = 0; v < 8; ++v) {
    int row = ci + v + 8 * h;                    // D: M = v + 8h
    atomicAdd(&out[(size_t)row * CCH + cj + m], acc[v]);
  }
}

// ---------------------------------------------------------------------------
// pf_next = (pf - mu) @ G + bias, with G supplied TRANSPOSED (Gt[col][k]).
// 256 threads = 8 waves.  Whole Gt staged column-major in LDS (stride 260,
// conflict-free, 8B-aligned for ds_load_b64) via async load-to-LDS when
// available; 32-row centered stripe also in LDS.  Each wave register-blocks a
// 2x2 tile grid: 4 ds_load_b64 -> 4 WMMAs per k-step, frags reused twice.
// ---------------------------------------------------------------------------
#define GST 260
__global__ void k_apply(const float* __restrict__ src, const float* __restrict__ mu,
                        const float* __restrict__ Gt, const float* __restrict__ bias,
                        float* __restrict__ dst) {
  extern __shared__ float lds[];
  float* gs = lds;              // G^T column-major: gs[col*GST + k]
  float* as = lds + 256 * GST;  // centered rows:    as[r*GST + c]
  int t = threadIdx.x;
  size_t row0 = (size_t)blockIdx.x * 32;

  // ---- stage G^T: each 16B global chunk = 4 consecutive k of one column ----
#if HAVE_ASYNC_LDS
  for (int i = t; i < 16384; i += 256) {
    int col = i >> 6, k4 = (i & 63) << 2;
    __builtin_amdgcn_global_load_async_to_lds_b128(
        (gbl_i32x4*)(Gt + ((size_t)i << 2)),
        (lds_i32x4*)(gs + col * GST + k4),
        0, 0);
  }
#else
  {
    const float4* g4 = (const float4*)Gt;
    for (int i = t; i < 16384; i += 256) {
      int col = i >> 6, k4 = (i & 63) << 2;
      *(float4*)(gs + col * GST + k4) = g4[i];
    }
  }
#endif

  // ---- stage centered row stripe ----
  for (int i = t; i < 32 * 256; i += 256) {
    int r = i >> 8, c = i & 255;
    as[r * GST + c] = src[(row0 + r) * CCH + c] - mu[c];
  }
  // prefetch next stripe (-> global_prefetch_b8)
  if (blockIdx.x + 1 < gridDim.x)
    __builtin_prefetch(&src[(row0 + 32) * CCH + (size_t)t * 4], 0, 0);

#if HAVE_ASYNC_LDS
#if HAVE_WAIT_ASYNC
  __builtin_amdgcn_s_wait_asynccnt(0);
#else
  asm volatile("s_wait_asynccnt 0" ::: "memory");
#endif
#endif
  __syncthreads();

  int w = t >> 5;               // wave id 0..7 -> col tiles {2w, 2w+1}
  int l = t & 31;
  int m = l & 15, h = l >> 4;
  int cb0 = w * 32, cb1 = cb0 + 16;

  v8f acc00 = {}, acc01 = {}, acc10 = {}, acc11 = {};
  const float* a0 = as + (size_t)m * GST;          // row tile 0, A row m
  const float* a1 = as + (size_t)(16 + m) * GST;   // row tile 1
  const float* b0 = gs + (size_t)(cb0 + m) * GST;  // col tile 0, B col n=m
  const float* b1 = gs + (size_t)(cb1 + m) * GST;  // col tile 1

  for (int k = 0; k < 256; k += 4) {
    int kk = k + 2 * h;                 // kk always even -> 8B aligned
    v2f av0 = *(const v2f*)(a0 + kk);   // A[m][kk], A[m][kk+1]
    v2f av1 = *(const v2f*)(a1 + kk);
    v2f bv0 = *(const v2f*)(b0 + kk);   // B[kk][n], B[kk+1][n]
    v2f bv1 = *(const v2f*)(b1 + kk);
    acc00 = wmma4(av0, bv0, acc00);
    acc01 = wmma4(av0, bv1, acc01);
    acc10 = wmma4(av1, bv0, acc10);
    acc11 = wmma4(av1, bv1, acc11);
  }

  float bb0 = bias[cb0 + m], bb1 = bias[cb1 + m];
#pragma unroll
  for (int v = 0; v < 8; ++v) {
    size_t r0 = row0 + v + 8 * h;        // D: M = v + 8h
    size_t r1 = row0 + 16 + v + 8 * h;
    dst[r0 * CCH + cb0 + m] = acc00[v] + bb0;
    dst[r0 * CCH + cb1 + m] = acc01[v] + bb1;
    dst[r1 * CCH + cb0 + m] = acc10[v] + bb0;
    dst[r1 * CCH + cb1 + m] = acc11[v] + bb1;
  }
}

// ---------------------------------------------------------------------------
// host orchestration
// ---------------------------------------------------------------------------
extern "C" void kernel_launch(void* const* d_in, const int* in_sizes, int n_in,
                              void* d_out, int out_size, void* d_ws, size_t ws_size,
                              hipStream_t stream) {
  (void)n_in; (void)out_size; (void)ws_size;
  const float* pastiche = (const float*)d_in[0];
  const float* style    = (const float*)d_in[1];
  const int ITERS = 64;                 // reference setup value
  const int N  = in_sizes[0] / CCH;     // 65536
  const int NE = N * CCH;

  float* ws = (float*)d_ws;
  size_t off = 0;
  float* buf1 = ws + off; off += (size_t)NE;
  float* mu_p = ws + off; off += CCH;
  float* mu_s = ws + off; off += CCH;
  float* Ar   = ws + off; off += CCH * CCH;
  float* Rm   = ws + off; off += CCH * CCH;
  float* Cp   = ws + off; off += CCH * CCH;
  float* Cs   = ws + off; off += CCH * CCH;
  float* covT = ws + off; off += CCH * CCH;
  float* covS = ws + off; off += CCH * CCH;
  float* Amap = ws + off; off += CCH * CCH;
  float* T1   = ws + off; off += CCH * CCH;
  float* Gtm  = ws + off; off += CCH * CCH;   // G transposed
  float* tmp  = ws + off; off += CCH * CCH;

  const int KS = 64;
  const int chunk = N / KS;             // 1024
  const size_t qrLds    = (size_t)(256 * QS + 256) * sizeof(float);
  const size_t cholLds  = (size_t)(256 * 257) * sizeof(float);
  const size_t applyLds = (size_t)(256 * GST + 32 * GST) * sizeof(float);

  // ---- style statistics (iteration-invariant) ----
  k_zero  <<<1, 256, 0, stream>>>(mu_s, CCH);
  k_colsum<<<N / 256, 256, 0, stream>>>(style, mu_s);
  k_scale <<<1, 256, 0, stream>>>(mu_s, 1.0f / (float)N, CCH);
  k_zero  <<<CCH, 256, 0, stream>>>(Cs, CCH * CCH);
  k_cov   <<<dim3(16, 16, KS), 32, 0, stream>>>(style, mu_s, Cs, chunk);
  k_scale <<<CCH, 256, 0, stream>>>(Cs, 1.0f / (float)N, CCH * CCH);

  float* cur = (float*)d_out;
  float* alt = buf1;
  k_copy<<<(NE + 255) / 256, 256, 0, stream>>>(cur, pastiche, NE);

  for (int it = 0; it < ITERS; ++it) {
    // Haar rotation R
    k_gauss<<<CCH, 256, 0, stream>>>(Ar, 0x9E3779B9u * (unsigned)(it + 1));
    k_qr<<<1, 256, qrLds, stream>>>(Ar, Rm);

    // pastiche mean + covariance (unrotated)
    k_zero  <<<1, 256, 0, stream>>>(mu_p, CCH);
    k_colsum<<<N / 256, 256, 0, stream>>>(cur, mu_p);
    k_scale <<<1, 256, 0, stream>>>(mu_p, 1.0f / (float)N, CCH);
    k_zero  <<<CCH, 256, 0, stream>>>(Cp, CCH * CCH);
    k_cov   <<<dim3(16, 16, KS), 32, 0, stream>>>(cur, mu_p, Cp, chunk);
    k_scale <<<CCH, 256, 0, stream>>>(Cp, 1.0f / (float)N, CCH * CCH);

    // rotated covariances: cov_t = R^T Cp R + eps I ; cov_s = R^T Cs R + eps I
    k_mm256<<<256, 256, 0, stream>>>(Cp, Rm, tmp, 0, 0, 0.0f);
    k_mm256<<<256, 256, 0, stream>>>(Rm, tmp, covT, 1, 0, EPSR);
    k_mm256<<<256, 256, 0, stream>>>(Cs, Rm, tmp, 0, 0, 0.0f);
    k_mm256<<<256, 256, 0, stream>>>(Rm, tmp, covS, 1, 0, EPSR);

    // Lt, Ls, Amap = Ls * inv(Lt)
    k_chol<<<1, 256, cholLds, stream>>>(covT);
    k_chol<<<1, 256, cholLds, stream>>>(covS);
    k_trisolve<<<1, 256, cholLds, stream>>>(covT, covS, Amap);

    // G = R @ Amap^T @ R^T; produce G TRANSPOSED: Gt = R @ T1^T with
    // T1 = R @ Amap^T  =>  Gt[i][j] = sum_k R[i][k] * T1[j][k]
    k_mm256<<<256, 256, 0, stream>>>(Rm, Amap, T1, 0, 1, 0.0f);
    k_mm256<<<256, 256, 0, stream>>>(Rm, T1, Gtm, 0, 1, 0.0f);

    // pf_next = (pf - mu_p) @ G + mu_style
    k_apply<<<N / 32, 256, applyLds, stream>>>(cur, mu_p, Gtm, mu_s, alt);
    float* t2 = cur; cur = alt; alt = t2;
  }
  if (cur != (float*)d_out)
    k_copy<<<(NE + 255) / 256, 256, 0, stream>>>((float*)d_out, cur, NE);
}